// Loss_58317065945194
// MI455X (gfx1250) — compile-verified
//
#include <hip/hip_runtime.h>
#include <hip/hip_bf16.h>
#include <math.h>

typedef float v2f __attribute__((ext_vector_type(2)));
typedef float v8f __attribute__((ext_vector_type(8)));

static constexpr int C_DIM   = 10;  // channels per row (reference: C=10)
static constexpr int NCHUNKS = 3;   // K chunks of 4 cover K=0..11 (>=10 is zero padding)

// ---------------------------------------------------------------------------
// Stage 1: each wave processes 32 rows/iteration.
//   cdf = L (16x16 lower-tri ones, rows>=10 zeroed) x diff^T  via 3 chained
//   V_WMMA_F32_16X16X4_F32 per 16-row tile (2 tiles per wave iteration).
//   per_sample = (mean_i |cdf_i|^r)^(1/r); block-partial sums -> d_ws.
// ---------------------------------------------------------------------------
__global__ __launch_bounds__(256) void emd_stage1(const float* __restrict__ p,
                                                  const float* __restrict__ q,
                                                  const int* __restrict__ rptr,
                                                  float* __restrict__ partials,
                                                  int nrows)
{
    const int lane          = threadIdx.x & 31;
    const int wave          = threadIdx.x >> 5;
    const int wavesPerBlock = blockDim.x >> 5;                     // 8
    const long long globalWave = (long long)blockIdx.x * wavesPerBlock + wave;
    const long long waveStride = (long long)gridDim.x * wavesPerBlock * 32;

    const int   r    = *rptr;          // uniform scalar (setup: r = 2)
    const float rf   = (float)r;
    const bool  hi   = (lane >= 16);
    const int   m    = lane & 15;
    const float invC = 1.0f / (float)C_DIM;

    // A-matrix (16x4 f32) per K-chunk c: A[M,K] = L[M, 4c+K],
    // L[i,j] = (j <= i && i < C_DIM) ? 1 : 0.
    // Layout (ISA 7.12.2): lanes hold M = lane&15; VGPR0 = K {0|2}, VGPR1 = K {1|3}.
    v2f A[NCHUNKS];
#pragma unroll
    for (int c = 0; c < NCHUNKS; ++c) {
        const int k0 = 4 * c + (hi ? 2 : 0);
        const int k1 = k0 + 1;
        A[c].x = (k0 <= m && m < C_DIM) ? 1.0f : 0.0f;
        A[c].y = (k1 <= m && m < C_DIM) ? 1.0f : 0.0f;
    }

    float lacc = 0.0f;

    for (long long base = globalWave * 32; base < nrows; base += waveStride) {
        // Load this lane's row (rows are 40B apart -> 8B-aligned b64 loads).
        float diff[C_DIM];
        if (base + 32 <= (long long)nrows) {
            // Fast path: whole 32-row tile in range (wave-uniform, no selects).
            const float* pr = p + (base + lane) * C_DIM;
            const float* qr = q + (base + lane) * C_DIM;
#pragma unroll
            for (int k = 0; k < C_DIM; k += 2) {
                const float2 pv = *(const float2*)(pr + k);
                const float2 qv = *(const float2*)(qr + k);
                diff[k]     = pv.x - qv.x;
                diff[k + 1] = pv.y - qv.y;
            }
        } else {
            const long long row = base + lane;
#pragma unroll
            for (int k = 0; k < C_DIM; ++k) diff[k] = 0.0f;
            if (row < nrows) {
                const float* pr = p + row * C_DIM;
                const float* qr = q + row * C_DIM;
#pragma unroll
                for (int k = 0; k < C_DIM; k += 2) {
                    const float2 pv = *(const float2*)(pr + k);
                    const float2 qv = *(const float2*)(qr + k);
                    diff[k]     = pv.x - qv.x;
                    diff[k + 1] = pv.y - qv.y;
                }
            }
        }
        // Prefetch next sweep (global_prefetch_b8; speculative).
        if (base + waveStride < nrows) {
            __builtin_prefetch(p + (base + waveStride + lane) * C_DIM, 0, 0);
            __builtin_prefetch(q + (base + waveStride + lane) * C_DIM, 0, 0);
        }

        // Broadcast rows into B-operand staging: tile0 = rows base..base+15,
        // tile1 = rows base+16..base+31. Every lane needs all channels of
        // row (lane&15) [+16 for tile1] -> ds_bpermute.
        float t0[12], t1[12];
#pragma unroll
        for (int k = 0; k < C_DIM; ++k) {
            t0[k] = __shfl(diff[k], m, 32);
            t1[k] = __shfl(diff[k], 16 + m, 32);
        }
#pragma unroll
        for (int k = C_DIM; k < 12; ++k) { t0[k] = 0.0f; t1[k] = 0.0f; }

        // D = sum_c A_c x B_c : chained K=4 WMMAs, full f32 precision.
        v8f acc0 = {0.f,0.f,0.f,0.f,0.f,0.f,0.f,0.f};
        v8f acc1 = {0.f,0.f,0.f,0.f,0.f,0.f,0.f,0.f};
#pragma unroll
        for (int c = 0; c < NCHUNKS; ++c) {
            v2f b0, b1;                       // B (4x16): VGPR0 = K{0|2}, VGPR1 = K{1|3}
            b0.x = hi ? t0[4*c + 2] : t0[4*c + 0];
            b0.y = hi ? t0[4*c + 3] : t0[4*c + 1];
            b1.x = hi ? t1[4*c + 2] : t1[4*c + 0];
            b1.y = hi ? t1[4*c + 3] : t1[4*c + 1];
            acc0 = __builtin_amdgcn_wmma_f32_16x16x4_f32(false, A[c], false, b0,
                                                         (short)0, acc0, false, false);
            acc1 = __builtin_amdgcn_wmma_f32_16x16x4_f32(false, A[c], false, b1,
                                                         (short)0, acc1, false, false);
        }

        // D layout: lane N = batch col, VGPR v = row i (lanes<16) / i+8 (lanes>=16).
        float s0 = 0.0f, s1 = 0.0f;
        if (r == 2) {
#pragma unroll
            for (int v = 0; v < 8; ++v) { s0 += acc0[v]*acc0[v]; s1 += acc1[v]*acc1[v]; }
        } else {
            // Fallback (unused for setup's r=2): fast-math |x|^r via v_log/v_exp.
#pragma unroll
            for (int v = 0; v < 8; ++v) {
                s0 += __expf(rf * __logf(fabsf(acc0[v])));
                s1 += __expf(rf * __logf(fabsf(acc1[v])));
            }
        }
        const float tot0 = s0 + __shfl_xor(s0, 16, 32);   // combine i=0..7 with i=8..15
        const float tot1 = s1 + __shfl_xor(s1, 16, 32);
        float ps0, ps1;
        if (r == 2) {
            ps0 = sqrtf(tot0 * invC);
            ps1 = sqrtf(tot1 * invC);
        } else {
            ps0 = __expf(__logf(tot0 * invC) * (1.0f / rf));
            ps1 = __expf(__logf(tot1 * invC) * (1.0f / rf));
        }
        if (!hi) lacc += ps0 + ps1;   // lanes l and l^16 hold identical values; count once
    }

    // Deterministic block reduction (fixed tree, no atomics).
#pragma unroll
    for (int off = 16; off >= 1; off >>= 1) lacc += __shfl_xor(lacc, off, 32);
    __shared__ float sdata[8];
    if (lane == 0) sdata[wave] = lacc;
    __syncthreads();
    if (threadIdx.x == 0) {
        float s = 0.0f;
        for (int w = 0; w < wavesPerBlock; ++w) s += sdata[w];
        partials[blockIdx.x] = s;
    }
}

// ---------------------------------------------------------------------------
// Stage 2: single-block deterministic reduction of block partials, scale 1/B.
// ---------------------------------------------------------------------------
__global__ __launch_bounds__(256) void emd_stage2(const float* __restrict__ partials,
                                                  int n, float* __restrict__ out,
                                                  float invB)
{
    float s = 0.0f;
    for (int i = threadIdx.x; i < n; i += 256) s += partials[i];
#pragma unroll
    for (int off = 16; off >= 1; off >>= 1) s += __shfl_xor(s, off, 32);
    __shared__ float sd[8];
    const int wave = threadIdx.x >> 5;
    const int lane = threadIdx.x & 31;
    if (lane == 0) sd[wave] = s;
    __syncthreads();
    if (threadIdx.x == 0) {
        float t = 0.0f;
        for (int w = 0; w < 8; ++w) t += sd[w];
        out[0] = t * invB;
    }
}

extern "C" void kernel_launch(void* const* d_in, const int* in_sizes, int n_in,
                              void* d_out, int out_size, void* d_ws, size_t ws_size,
                              hipStream_t stream)
{
    (void)n_in; (void)out_size;
    const float* p    = (const float*)d_in[0];
    const float* q    = (const float*)d_in[1];
    const int*   rptr = (const int*)d_in[2];

    const int nrows = in_sizes[0] / C_DIM;      // B = 2,097,152

    int blocks = 4096;                          // 32768 waves; 2 grid-stride sweeps
    if (ws_size < (size_t)blocks * sizeof(float))
        blocks = (int)(ws_size / sizeof(float));
    if (blocks < 1) blocks = 1;

    float* partials = (float*)d_ws;
    emd_stage1<<<blocks, 256, 0, stream>>>(p, q, rptr, partials, nrows);
    emd_stage2<<<1, 256, 0, stream>>>(partials, blocks, (float*)d_out,
                                      1.0f / (float)nrows);
}